// GATConv2d_60997125538362
// MI455X (gfx1250) — compile-verified
//
#include <hip/hip_runtime.h>
#include <hip/hip_bf16.h>
#include <math.h>

// Problem constants (from the reference)
#define BATCH   4
#define C_IN    128
#define C_OUT   128
#define NNODES  32768
#define KNBR    9
#define NEG_SLOPE 0.2f

typedef __attribute__((ext_vector_type(2))) float v2f;
typedef __attribute__((ext_vector_type(8))) float v8f;

// ---------------------------------------------------------------------------
// Kernel 1: h[b, n, o] = relu( sum_c W[o,c] * x[b,c,n] )
// Stored node-major (b, n, o) so each node's 128 channels are one contiguous
// 512B line -> the neighbor gather in kernel 2 becomes coalesced.
//
// One wave computes a 16(node) x 128(all C_OUT) strip with 8 accumulators,
// looping K = C_IN in steps of 4 using V_WMMA_F32_16X16X4_F32.
//
// A fragment (16x4, f32): lanes 0-15 hold rows M=lane, K=0 (v.x) / K=1 (v.y);
//                         lanes 16-31 hold rows M=lane-16, K=2 / K=3.
// B fragment (4x16, f32): lanes 0-15 hold cols N=lane, K=0/1;
//                         lanes 16-31 hold cols N=lane-16, K=2/3.
// C/D (16x16, f32): VGPR v -> row v (lanes 0-15) / row v+8 (lanes 16-31),
//                   col = lane & 15.
// ---------------------------------------------------------------------------
__global__ __launch_bounds__(128) void gat_gemm_relu_kernel(
    const float* __restrict__ x,   // (B, C_IN, N)
    const float* __restrict__ W,   // (C_OUT, C_IN)
    float* __restrict__ h)         // (B, N, C_OUT) workspace
{
    const int lane = threadIdx.x & 31;
    const int wave = threadIdx.x >> 5;
    const int b    = blockIdx.y;
    const int n0   = (blockIdx.x * 4 + wave) * 16;   // node tile base

    const float* X = x + (size_t)b * C_IN * NNODES;

    const int m  = lane & 15;        // row within tile (A) / col (B, C/D)
    const int kk = (lane >> 4) * 2;  // K sub-offset for this lane half

    v8f acc[8];
#pragma unroll
    for (int i = 0; i < 8; ++i)
        acc[i] = (v8f){0.f, 0.f, 0.f, 0.f, 0.f, 0.f, 0.f, 0.f};

    for (int k0 = 0; k0 < C_IN; k0 += 4) {
        // A fragment: x[b, k0+kk(+1), n0+m]  (16 contiguous lanes per row)
        v2f a;
        a.x = X[(size_t)(k0 + kk)     * NNODES + n0 + m];
        a.y = X[(size_t)(k0 + kk + 1) * NNODES + n0 + m];

#pragma unroll
        for (int ot = 0; ot < 8; ++ot) {
            // B fragment: W[ot*16 + m, k0+kk(+1)]  (W row-major over C_IN)
            const float* wrow = W + (size_t)(ot * 16 + m) * C_IN + k0 + kk;
            v2f bf;
            bf.x = wrow[0];
            bf.y = wrow[1];
            acc[ot] = __builtin_amdgcn_wmma_f32_16x16x4_f32(
                /*neg_a=*/false, a, /*neg_b=*/false, bf,
                /*c_mod=*/(short)0, acc[ot],
                /*reuse_a=*/false, /*reuse_b=*/false);
        }
    }

    // Epilogue: relu + store to (b, n, o)
    const int col   = lane & 15;
    const int rbase = (lane >> 4) * 8;
#pragma unroll
    for (int ot = 0; ot < 8; ++ot) {
#pragma unroll
        for (int v = 0; v < 8; ++v) {
            const int row = v + rbase;
            float val = acc[ot][v];
            val = val > 0.f ? val : 0.f;
            h[((size_t)b * NNODES + n0 + row) * C_OUT + ot * 16 + col] = val;
        }
    }
}

// ---------------------------------------------------------------------------
// Kernel 2: per-node attention + gather-max.
// 1024 threads = 32 waves; wave w handles node g = blockIdx.x*32 + w.
// Lane l owns channels 4l..4l+3 (float4). Neighbor columns are contiguous
// 512B reads thanks to the (b,n,c) layout of h (L2-resident: 64MB < 192MB).
// Output is transposed through padded LDS so the (B,C,N) store coalesces.
// ---------------------------------------------------------------------------
__global__ __launch_bounds__(1024) void gat_attn_gather_kernel(
    const float* __restrict__ h,     // (B, N, C_OUT)
    const int*   __restrict__ edge,  // edge_index[0] flat: (B, N, K)
    const float* __restrict__ att,   // (2*C_OUT)
    const float* __restrict__ bias,  // (C_OUT)
    float* __restrict__ out)         // (B, C_OUT, N)
{
    __shared__ float lds[32 * 132];  // 32 nodes x 128 ch, +4 pad per row

    const int lane = threadIdx.x & 31;
    const int wave = threadIdx.x >> 5;
    const int g    = blockIdx.x * 32 + wave;  // global node id in [0, B*N)
    const int b    = g >> 15;                 // g / NNODES (N = 2^15)

    // Neighbor indices for this node: lane k<9 loads one, broadcast later.
    int myidx = 0;
    if (lane < KNBR) myidx = edge[(size_t)g * KNBR + lane];

    const float4 a_i = *(const float4*)(att + 4 * lane);
    const float4 a_j = *(const float4*)(att + C_OUT + 4 * lane);

    // Self score s_i = a_i . h[b,:,n]
    const float4 self = *(const float4*)(h + (size_t)g * C_OUT + 4 * lane);
    float si = self.x * a_i.x + self.y * a_i.y + self.z * a_i.z + self.w * a_i.w;
#pragma unroll
    for (int off = 16; off > 0; off >>= 1) si += __shfl_xor(si, off, 32);

    // Gather neighbor columns; compute raw attention logits.
    float4 cols[KNBR];
    float  alpha[KNBR];
    const size_t hb = (size_t)b * NNODES;
#pragma unroll
    for (int k = 0; k < KNBR; ++k) {
        const int nk = __shfl(myidx, k, 32);
        const float4 c = *(const float4*)(h + (hb + nk) * C_OUT + 4 * lane);
        cols[k] = c;
        float sj = c.x * a_j.x + c.y * a_j.y + c.z * a_j.z + c.w * a_j.w;
#pragma unroll
        for (int off = 16; off > 0; off >>= 1) sj += __shfl_xor(sj, off, 32);
        const float a = si + sj;
        alpha[k] = a > 0.f ? a : NEG_SLOPE * a;   // leaky relu
    }

    // Softmax over K=9 (redundant per lane, cheap).
    float mx = alpha[0];
#pragma unroll
    for (int k = 1; k < KNBR; ++k) mx = fmaxf(mx, alpha[k]);
    float den = 0.f;
#pragma unroll
    for (int k = 0; k < KNBR; ++k) { alpha[k] = __expf(alpha[k] - mx); den += alpha[k]; }
    const float inv = 1.f / den;

    // out_c = max_k cols[k].c * alpha[k]  (all terms >= 0 after relu/softmax)
    float4 o;
    o.x = cols[0].x * (alpha[0] * inv);
    o.y = cols[0].y * (alpha[0] * inv);
    o.z = cols[0].z * (alpha[0] * inv);
    o.w = cols[0].w * (alpha[0] * inv);
#pragma unroll
    for (int k = 1; k < KNBR; ++k) {
        const float ak = alpha[k] * inv;
        o.x = fmaxf(o.x, cols[k].x * ak);
        o.y = fmaxf(o.y, cols[k].y * ak);
        o.z = fmaxf(o.z, cols[k].z * ak);
        o.w = fmaxf(o.w, cols[k].w * ak);
    }
    const float4 bs = *(const float4*)(bias + 4 * lane);
    o.x += bs.x; o.y += bs.y; o.z += bs.z; o.w += bs.w;

    // Transpose through LDS (padded stride 132 floats avoids bank conflicts).
    float* row = lds + wave * 132 + 4 * lane;
    row[0] = o.x; row[1] = o.y; row[2] = o.z; row[3] = o.w;
    __syncthreads();

    // Coalesced store: thread t -> channel c = t>>3, nodes nn..nn+3.
    const int n0 = (blockIdx.x * 32) & (NNODES - 1);  // node base within batch
    const int c  = threadIdx.x >> 3;
    const int nn = (threadIdx.x & 7) * 4;
    float4 res;
    res.x = lds[(nn + 0) * 132 + c];
    res.y = lds[(nn + 1) * 132 + c];
    res.z = lds[(nn + 2) * 132 + c];
    res.w = lds[(nn + 3) * 132 + c];
    *(float4*)(out + ((size_t)(b * C_OUT + c) * NNODES) + n0 + nn) = res;
}

// ---------------------------------------------------------------------------
// kernel_launch: inputs in setup_inputs() order: x, W, att, bias, edge_index
// ---------------------------------------------------------------------------
extern "C" void kernel_launch(void* const* d_in, const int* in_sizes, int n_in,
                              void* d_out, int out_size, void* d_ws, size_t ws_size,
                              hipStream_t stream)
{
    const float* x    = (const float*)d_in[0];   // (B, C_IN, N, 1)
    const float* W    = (const float*)d_in[1];   // (C_OUT, C_IN)
    const float* att  = (const float*)d_in[2];   // (2*C_OUT,)
    const float* bias = (const float*)d_in[3];   // (C_OUT,)
    const int*   edge = (const int*)d_in[4];     // (2, B, N, K); we use [0]
    float*       out  = (float*)d_out;           // (B, C_OUT, N, 1)

    float* h = (float*)d_ws;                     // (B, N, C_OUT) = 64 MB

    // Kernel 1: WMMA GEMM + relu.  grid: (N/16/4 waves, B), 4 waves/block.
    dim3 g1((NNODES / 16) / 4, BATCH);
    gat_gemm_relu_kernel<<<g1, 128, 0, stream>>>(x, W, h);

    // Kernel 2: attention gather.  32 nodes per 1024-thread block.
    dim3 g2((BATCH * NNODES) / 32);
    gat_attn_gather_kernel<<<g2, 1024, 0, stream>>>(h, edge, att, bias, out);
}